// UniGINConv_82128364634686
// MI455X (gfx1250) — compile-verified
//
#include <hip/hip_runtime.h>
#include <hip/hip_bf16.h>

// UniGIN hypergraph convolution for MI455X (gfx1250), wave32.
//
// Pipeline:
//   1) Xp = X @ W           -- f32 WMMA (v_wmma_f32_16x16x4_f32), 16x16 tile/wave
//   2) Xe = segsum_e(Xp[v]) -- gather + global_atomic_add_f32 (resolves in 192MB L2)
//   3) Xv = segsum_v(Xe[e]) -- gather + global_atomic_add_f32
//   4) out = gelu_exact((1+eps)*Xp + Xv)
//
// Problem constants (from reference):
#define N_NODES 100000
#define N_EDGES 50000
#define NNZ     1600000
#define DIM     128

typedef __attribute__((ext_vector_type(2))) float v2f;
typedef __attribute__((ext_vector_type(8))) float v8f;

// ---------------------------------------------------------------------------
// 1) Dense GEMM: Xp[100000,128] = X[100000,128] @ W[128,128]
//    grid.x = 6250 (row tiles of 16), block = 256 (8 waves).
//    Wave w computes the 16x16 tile at (blockIdx.x*16, w*16).
// ---------------------------------------------------------------------------
__global__ void __launch_bounds__(256)
unigin_gemm_wmma_f32(const float* __restrict__ X,
                     const float* __restrict__ W,
                     float* __restrict__ Xp) {
    const int wave = threadIdx.x >> 5;
    const int lane = threadIdx.x & 31;
    const int m    = lane & 15;       // A: row in tile / B: col in tile / D: N col
    const int grp  = lane >> 4;       // half-wave selects K-pair (A/B) & +8 rows (D)

    const int row0 = blockIdx.x * 16;
    const int col0 = wave * 16;

    v8f acc = {};
    const float* __restrict__ arow = X + (size_t)(row0 + m) * DIM;

    #pragma unroll 4
    for (int k0 = 0; k0 < DIM; k0 += 4) {
        const int ka = k0 + grp * 2;
        // A 16x4 f32: VGPR0 = K=ka, VGPR1 = K=ka+1 (contiguous -> b64 load)
        v2f a;
        a.x = arow[ka + 0];
        a.y = arow[ka + 1];
        // B 4x16 f32: same striping, lane%16 = N column
        v2f b;
        b.x = W[(size_t)(ka + 0) * DIM + col0 + m];
        b.y = W[(size_t)(ka + 1) * DIM + col0 + m];
        acc = __builtin_amdgcn_wmma_f32_16x16x4_f32(
            /*neg_a=*/false, a, /*neg_b=*/false, b,
            /*c_mod=*/(short)0, acc, /*reuse_a=*/false, /*reuse_b=*/false);
    }

    // D 16x16 f32: VGPR v, lanes 0-15 -> M=v, lanes 16-31 -> M=v+8; N = lane%16
    #pragma unroll
    for (int v = 0; v < 8; ++v) {
        Xp[(size_t)(row0 + v + grp * 8) * DIM + col0 + m] = acc[v];
    }
}

// ---------------------------------------------------------------------------
// 2)/3) Incidence gather + segment-sum via f32 atomics.
//    One wave per nnz entry: 32 lanes x float4 = full 128-float (512B) row.
//    dst[scatter_idx[i]] += src[gather_idx[i]]
// ---------------------------------------------------------------------------
__global__ void __launch_bounds__(256)
unigin_gather_scatter_add(const float* __restrict__ src,
                          float* __restrict__ dst,
                          const int* __restrict__ gather_idx,
                          const int* __restrict__ scatter_idx,
                          int nnz) {
    const int gwave = (int)((blockIdx.x * blockDim.x + threadIdx.x) >> 5);
    const int lane  = threadIdx.x & 31;
    if (gwave >= nnz) return;

    const int g = gather_idx[gwave];
    const int s = scatter_idx[gwave];

    const float4 v = ((const float4*)(src + (size_t)g * DIM))[lane];
    float* d = dst + (size_t)s * DIM + lane * 4;
    atomicAdd(d + 0, v.x);
    atomicAdd(d + 1, v.y);
    atomicAdd(d + 2, v.z);
    atomicAdd(d + 3, v.w);
}

// ---------------------------------------------------------------------------
// 4) out = gelu_exact((1+eps)*Xp + Xv), float4 per thread.
// ---------------------------------------------------------------------------
__device__ __forceinline__ float gelu_erf(float x) {
    return 0.5f * x * (1.0f + erff(x * 0.70710678118654752440f));
}

__global__ void __launch_bounds__(256)
unigin_combine_gelu(const float* __restrict__ Xp,
                    const float* __restrict__ Xv,
                    const float* __restrict__ eps,
                    float* __restrict__ out,
                    int total4) {
    const int i = (int)(blockIdx.x * blockDim.x + threadIdx.x);
    if (i >= total4) return;
    const float e = 1.0f + eps[0];
    const float4 p = ((const float4*)Xp)[i];
    const float4 v = ((const float4*)Xv)[i];
    float4 r;
    r.x = gelu_erf(e * p.x + v.x);
    r.y = gelu_erf(e * p.y + v.y);
    r.z = gelu_erf(e * p.z + v.z);
    r.w = gelu_erf(e * p.w + v.w);
    ((float4*)out)[i] = r;
}

// ---------------------------------------------------------------------------
extern "C" void kernel_launch(void* const* d_in, const int* in_sizes, int n_in,
                              void* d_out, int out_size, void* d_ws, size_t ws_size,
                              hipStream_t stream) {
    const float* X      = (const float*)d_in[0];   // [100000,128]
    const float* W      = (const float*)d_in[1];   // [128,128]
    const float* eps    = (const float*)d_in[2];   // [1]
    const int*   vertex = (const int*)d_in[3];     // [NNZ]
    const int*   edges  = (const int*)d_in[4];     // [NNZ]
    (void)in_sizes; (void)n_in; (void)out_size; (void)ws_size;

    // Workspace layout (all 256B-aligned):
    //   Xp: N_NODES*DIM f32 = 51.2 MB
    //   Xe: N_EDGES*DIM f32 = 25.6 MB
    //   Xv: N_NODES*DIM f32 = 51.2 MB
    char* ws = (char*)d_ws;
    float* Xp = (float*)(ws);
    float* Xe = (float*)(ws + (size_t)N_NODES * DIM * sizeof(float));
    float* Xv = (float*)(ws + (size_t)(N_NODES + N_EDGES) * DIM * sizeof(float));

    // Zero accumulation buffers every call (harness poisons ws; atomics accumulate).
    hipMemsetAsync(Xe, 0, (size_t)N_EDGES * DIM * sizeof(float), stream);
    hipMemsetAsync(Xv, 0, (size_t)N_NODES * DIM * sizeof(float), stream);

    // 1) Xp = X @ W
    unigin_gemm_wmma_f32<<<N_NODES / 16, 256, 0, stream>>>(X, W, Xp);

    // 2) Xe[e] += Xp[v]  (one wave per incidence entry, 8 waves/block)
    const int nblocks_sc = (NNZ + 7) / 8;
    unigin_gather_scatter_add<<<nblocks_sc, 256, 0, stream>>>(Xp, Xe, vertex, edges, NNZ);

    // 3) Xv[v] += Xe[e]
    unigin_gather_scatter_add<<<nblocks_sc, 256, 0, stream>>>(Xe, Xv, edges, vertex, NNZ);

    // 4) out = gelu((1+eps)*Xp + Xv)
    const int total4 = N_NODES * DIM / 4;
    unigin_combine_gelu<<<(total4 + 255) / 256, 256, 0, stream>>>(Xp, Xv, eps, d_out ? (float*)d_out : Xp, total4);
}